// ScenarioFilter_46926812676857
// MI455X (gfx1250) — compile-verified
//
#include <hip/hip_runtime.h>
#include <hip/hip_bf16.h>

typedef __attribute__((ext_vector_type(2))) float v2f;
typedef __attribute__((ext_vector_type(8))) float v8f;

#define S_DIM 512
#define B_DIM 64
#define N_DIM 32
#define T_DIM 96
#define H_DIM 256
#define K_MODEL 10
#define K_RAND 10
#define NT 3072   // N_DIM * T_DIM
#define BS (B_DIM * S_DIM)  // 32768

// ---------------------------------------------------------------------------
// Kernel 1: Y_feature[b*512+s][t] = sum_n Y[s,b,n,t]
// Grid: 32768 blocks (sb = s*B + b), 96 threads (t)
// ---------------------------------------------------------------------------
__global__ void k_feature(const float* __restrict__ Y, float* __restrict__ Yf) {
    int sb = blockIdx.x;            // s*64 + b
    int s  = sb >> 6;
    int b  = sb & 63;
    int t  = threadIdx.x;           // 0..95
    const float* base = Y + (size_t)sb * NT;
    float acc = 0.0f;
#pragma unroll
    for (int n = 0; n < N_DIM; ++n) acc += base[n * T_DIM + t];
    Yf[(size_t)(b * S_DIM + s) * T_DIM + t] = acc;
}

// ---------------------------------------------------------------------------
// Kernel 2: fused MLP.  h = relu(Yf @ W1 + b1) via WMMA f32 16x16x4 (K=96),
// then logits = h @ W2 + b2 scattered to (B, 10, S).
// Grid: 2048 blocks (16 rows each), 512 threads = 16 waves (one 16-col tile per wave)
// ---------------------------------------------------------------------------
__global__ void k_mlp(const float* __restrict__ Yf,
                      const float* __restrict__ W1,
                      const float* __restrict__ b1,
                      const float* __restrict__ W2,
                      const float* __restrict__ b2,
                      float* __restrict__ logits) {
    __shared__ float sh_h[16][H_DIM + 1];   // padded: stride 257 -> conflict-free
    __shared__ float sW2[H_DIM * K_MODEL];  // [c][k]

    const int tid  = threadIdx.x;
    const int lane = tid & 31;
    const int wv   = tid >> 5;          // 0..15
    const int l16  = lane & 15;
    const int khalf = (lane >> 4) << 1; // 0 or 2
    const int row0 = blockIdx.x << 4;   // 16 rows per block
    const int col0 = wv << 4;           // 16 cols per wave

    // cooperative load of W2 into LDS
    for (int i = tid; i < H_DIM * K_MODEL; i += 512) sW2[i] = W2[i];

    // WMMA: D(16x16) = Yf_tile(16x96) @ W1_tile(96x16)
    v8f c = {};
    const float* arow = Yf + (size_t)(row0 + l16) * T_DIM;
#pragma unroll
    for (int kk = 0; kk < T_DIM; kk += 4) {
        v2f a, bb;
        a.x  = arow[kk + khalf];
        a.y  = arow[kk + khalf + 1];
        bb.x = W1[(kk + khalf)     * H_DIM + col0 + l16];
        bb.y = W1[(kk + khalf + 1) * H_DIM + col0 + l16];
        c = __builtin_amdgcn_wmma_f32_16x16x4_f32(false, a, false, bb,
                                                  (short)0, c, false, false);
    }

    // bias + relu -> LDS (D layout: vgpr i = row i (lanes 0-15) / i+8 (lanes 16-31))
    const int colw  = col0 + l16;
    const int rbase = (lane < 16) ? 0 : 8;
    const float bias = b1[colw];
#pragma unroll
    for (int i = 0; i < 8; ++i) {
        float v = c[i] + bias;
        sh_h[rbase + i][colw] = v > 0.0f ? v : 0.0f;
    }
    __syncthreads();

    // second GEMM: 16 rows x 10 outputs, K=256 (tiny; 160 threads)
    if (tid < 16 * K_MODEL) {
        int row = tid / K_MODEL;
        int k   = tid % K_MODEL;
        float acc = b2[k];
#pragma unroll 8
        for (int cc = 0; cc < H_DIM; ++cc) acc += sh_h[row][cc] * sW2[cc * K_MODEL + k];
        int bs = row0 + row;
        int b  = bs >> 9;
        int s  = bs & 511;
        logits[((size_t)(b * K_MODEL + k) << 9) + s] = acc;
    }
}

// ---------------------------------------------------------------------------
// Kernel 3: per (b,k): softmax -> eps-mix -> renorm -> log -> +gumbel -> softmax
// Grid: 640 blocks, 256 threads (2 s-elements each). tau == 1.
// ---------------------------------------------------------------------------
__global__ void k_softmax(const float* __restrict__ logits,
                          const float* __restrict__ gumbel,
                          float* __restrict__ A) {
    __shared__ float red[256];
    const int bk  = blockIdx.x;
    const int tid = threadIdx.x;
    const float* lrow = logits + (size_t)bk * S_DIM;
    const float* grow = gumbel + (size_t)bk * S_DIM;

    float l0 = lrow[tid], l1 = lrow[tid + 256];

    // max
    red[tid] = fmaxf(l0, l1); __syncthreads();
    for (int o = 128; o > 0; o >>= 1) { if (tid < o) red[tid] = fmaxf(red[tid], red[tid + o]); __syncthreads(); }
    float m = red[0]; __syncthreads();

    float e0 = expf(l0 - m), e1 = expf(l1 - m);
    red[tid] = e0 + e1; __syncthreads();
    for (int o = 128; o > 0; o >>= 1) { if (tid < o) red[tid] += red[tid + o]; __syncthreads(); }
    float inv_z = 1.0f / red[0]; __syncthreads();

    const float eps_u = 0.1f / (float)S_DIM;
    float p0 = 0.9f * e0 * inv_z + eps_u;
    float p1 = 0.9f * e1 * inv_z + eps_u;

    red[tid] = p0 + p1; __syncthreads();
    for (int o = 128; o > 0; o >>= 1) { if (tid < o) red[tid] += red[tid + o]; __syncthreads(); }
    float inv_z2 = 1.0f / red[0]; __syncthreads();
    p0 *= inv_z2; p1 *= inv_z2;

    float x0 = logf(fmaxf(p0, 1e-12f)) + grow[tid];
    float x1 = logf(fmaxf(p1, 1e-12f)) + grow[tid + 256];

    red[tid] = fmaxf(x0, x1); __syncthreads();
    for (int o = 128; o > 0; o >>= 1) { if (tid < o) red[tid] = fmaxf(red[tid], red[tid + o]); __syncthreads(); }
    float m2 = red[0]; __syncthreads();

    float a0 = expf(x0 - m2), a1 = expf(x1 - m2);
    red[tid] = a0 + a1; __syncthreads();
    for (int o = 128; o > 0; o >>= 1) { if (tid < o) red[tid] += red[tid + o]; __syncthreads(); }
    float inv_z3 = 1.0f / red[0];

    A[(size_t)bk * S_DIM + tid]       = a0 * inv_z3;
    A[(size_t)bk * S_DIM + tid + 256] = a1 * inv_z3;
}

// ---------------------------------------------------------------------------
// Kernel 4: Y_mix[k, b, nt] = sum_s A[b,k,s] * Y[s,b,nt]  via WMMA f32 16x16x4.
// Grid: 256 blocks (4 per b), 512 threads = 16 waves, 3 col-tiles per wave.
// Streams Y exactly once (bandwidth-dominant pass). Writes out rows k'=10..19.
// ---------------------------------------------------------------------------
__global__ void k_mix(const float* __restrict__ Y,
                      const float* __restrict__ A,
                      float* __restrict__ out) {
    __shared__ float sA[16][516];  // stride 516 -> 16 rows hit distinct banks

    const int b     = blockIdx.x >> 2;
    const int chunk = blockIdx.x & 3;
    const int tid   = threadIdx.x;
    const int lane  = tid & 31;
    const int wv    = tid >> 5;
    const int l16   = lane & 15;
    const int khalf = (lane >> 4) << 1; // 0 or 2

    // load A[b] (10x512) into LDS, pad rows 10..15 with zeros
    for (int i = tid; i < 16 * S_DIM; i += 512) {
        int r = i >> 9, sc = i & 511;
        sA[r][sc] = (r < K_MODEL) ? A[((size_t)b * K_MODEL + r) * S_DIM + sc] : 0.0f;
    }
    __syncthreads();

    const int tile0 = chunk * 48 + wv * 3;      // 192 tiles of 16 cols per b
    const int col0  = tile0 << 4;
    const float* __restrict__ srow = &sA[l16][khalf];

    v8f acc0 = {}, acc1 = {}, acc2 = {};
    for (int s = 0; s < S_DIM; s += 4) {
        v2f a;
        a.x = srow[s];
        a.y = srow[s + 1];
        const float* y0 = Y + ((size_t)((s + khalf)     * B_DIM + b)) * NT + col0 + l16;
        const float* y1 = Y + ((size_t)((s + khalf + 1) * B_DIM + b)) * NT + col0 + l16;
        v2f b0, b1v, b2v;
        b0.x  = y0[0];   b0.y  = y1[0];
        b1v.x = y0[16];  b1v.y = y1[16];
        b2v.x = y0[32];  b2v.y = y1[32];
        acc0 = __builtin_amdgcn_wmma_f32_16x16x4_f32(false, a, false, b0,  (short)0, acc0, false, false);
        acc1 = __builtin_amdgcn_wmma_f32_16x16x4_f32(false, a, false, b1v, (short)0, acc1, false, false);
        acc2 = __builtin_amdgcn_wmma_f32_16x16x4_f32(false, a, false, b2v, (short)0, acc2, false, false);
    }

    // D tile: vgpr i -> k = i (lanes 0-15) / i+8 (lanes 16-31), col = col + l16.
    const int rbase = (lane < 16) ? 0 : 8;
#pragma unroll
    for (int j = 0; j < 3; ++j) {
        const v8f* accp = (j == 0) ? &acc0 : (j == 1) ? &acc1 : &acc2;
        const int col = col0 + (j << 4) + l16;
#pragma unroll
        for (int i = 0; i < 8; ++i) {
            int k = rbase + i;
            if (k < K_MODEL)
                out[((size_t)(K_RAND + k) * B_DIM + b) * NT + col] = (*accp)[i];
        }
    }
}

// ---------------------------------------------------------------------------
// Kernel 5: Y_rand gather -> out rows k'=0..9.
// Grid: 640 blocks (b*10+k), 256 threads.
// ---------------------------------------------------------------------------
__global__ void k_rand(const float* __restrict__ Y,
                       const int* __restrict__ idx_rand,
                       float* __restrict__ out) {
    const int bk = blockIdx.x;
    const int b  = bk / K_RAND;
    const int k  = bk % K_RAND;
    const int idx = idx_rand[b * K_RAND + k];
    const float* src = Y   + ((size_t)idx * B_DIM + b) * NT;
    float*       dst = out + ((size_t)k   * B_DIM + b) * NT;
#pragma unroll
    for (int i = threadIdx.x; i < NT; i += 256) dst[i] = src[i];
}

// ---------------------------------------------------------------------------
extern "C" void kernel_launch(void* const* d_in, const int* in_sizes, int n_in,
                              void* d_out, int out_size, void* d_ws, size_t ws_size,
                              hipStream_t stream) {
    const float* Y        = (const float*)d_in[0];   // (S,B,N,T)
    const float* W1       = (const float*)d_in[1];   // (96,256)
    const float* b1       = (const float*)d_in[2];   // (256,)
    const float* W2       = (const float*)d_in[3];   // (256,10)
    const float* b2       = (const float*)d_in[4];   // (10,)
    const float* gumbel   = (const float*)d_in[5];   // (B,10,S)
    const int*   idx_rand = (const int*)d_in[6];     // (B,10) int32
    float* out = (float*)d_out;                      // (20,B,N,T)

    // workspace layout (floats): Yf | logits | A   (~15.2 MB total)
    float* Yf     = (float*)d_ws;                        // 32768*96
    float* logits = Yf + (size_t)BS * T_DIM;             // 64*10*512
    float* Aw     = logits + (size_t)B_DIM * K_MODEL * S_DIM;

    k_feature<<<BS, T_DIM, 0, stream>>>(Y, Yf);
    k_mlp<<<BS / 16, 512, 0, stream>>>(Yf, W1, b1, W2, b2, logits);
    k_softmax<<<B_DIM * K_MODEL, 256, 0, stream>>>(logits, gumbel, Aw);
    k_mix<<<B_DIM * 4, 512, 0, stream>>>(Y, Aw, out);
    k_rand<<<B_DIM * K_RAND, 256, 0, stream>>>(Y, idx_rand, out);
}